// NADE_16862041604503
// MI455X (gfx1250) — compile-verified
//
#include <hip/hip_runtime.h>
#include <hip/hip_bf16.h>
#include <math.h>

// Problem constants (match reference)
#define NB 128   // batch
#define ND 1024  // D
#define NH 512   // H
#define CS 128   // chunk size along D
#define NC (ND / CS)  // 8 chunks
#define JT 16    // j-tile for the reduction phase in nade_seq

typedef __attribute__((ext_vector_type(2)))  float    v2f;
typedef __attribute__((ext_vector_type(8)))  float    v8f;

// ---------------------------------------------------------------------------
// Kernel 1: chunk partial-sum GEMM  S[c][b,h] = sum_{j in chunk c} x[b,j]*W[h,j]
// One wave32 computes one 16x16 (b,h) tile for one chunk via WMMA f32 16x16x4.
// ---------------------------------------------------------------------------
__global__ __launch_bounds__(256)
void nade_chunk_gemm(const float* __restrict__ x,
                     const float* __restrict__ W,
                     float* __restrict__ S) {
    const int tilesB = NB / 16;                  // 8
    const int tilesH = NH / 16;                  // 32
    const int tilesPerChunk = tilesB * tilesH;   // 256
    const int totalTiles = NC * tilesPerChunk;   // 2048

    const int wavesPerBlock = blockDim.x >> 5;
    const int waveId = blockIdx.x * wavesPerBlock + (threadIdx.x >> 5);
    if (waveId >= totalTiles) return;            // uniform per wave

    const int lane = threadIdx.x & 31;
    const int m    = lane & 15;                  // row (A) / col (B) within tile
    const int half = lane >> 4;                  // 0: K low pair, 1: K high pair

    const int c   = waveId / tilesPerChunk;
    const int rem = waveId % tilesPerChunk;
    const int b0  = (rem / tilesH) * 16;
    const int h0  = (rem % tilesH) * 16;

    const float* __restrict__ xrow = x + (size_t)(b0 + m) * ND;  // A row b0+m
    const float* __restrict__ wrow = W + (size_t)(h0 + m) * ND;  // B col h0+m
    const int jbase = c * CS;

    v8f acc = {};  // 16x16 f32 accumulator

    // V_WMMA_F32_16X16X4_F32: A 16x4 (lane half -> K = 2*half + {0,1}),
    // B 4x16 mirrored, C/D 16x16 f32.
    for (int k = 0; k < CS; k += 4) {
        const int j = jbase + k + 2 * half;
        v2f a  = { xrow[j], xrow[j + 1] };
        v2f bm = { wrow[j], wrow[j + 1] };
        acc = __builtin_amdgcn_wmma_f32_16x16x4_f32(
            false, a, false, bm, (short)0, acc, false, false);
    }

    // C/D layout: VGPR v -> M = v + 8*half, N = m
    float* __restrict__ Sc = S + (size_t)c * NB * NH;
#pragma unroll
    for (int v = 0; v < 8; ++v) {
        const int M = v + 8 * half;
        Sc[(size_t)(b0 + M) * NH + (h0 + m)] = acc[v];
    }
}

// ---------------------------------------------------------------------------
// Kernel 2: exclusive prefix over chunks (in place):  S[c] <- sum_{c'<c} S[c']
// ---------------------------------------------------------------------------
__global__ __launch_bounds__(256)
void nade_prefix(float* __restrict__ S) {
    const int idx = blockIdx.x * blockDim.x + threadIdx.x;
    if (idx >= NB * NH) return;
    float run = 0.0f;
#pragma unroll
    for (int c = 0; c < NC; ++c) {
        const float v = S[(size_t)c * NB * NH + idx];
        S[(size_t)c * NB * NH + idx] = run;
        run += v;
    }
}

__device__ __forceinline__ float wave_reduce_add(float v) {
#pragma unroll
    for (int off = 16; off > 0; off >>= 1)
        v += __shfl_xor(v, off, 32);
    return v;
}

// ---------------------------------------------------------------------------
// Kernel 3: sequential recurrence within a chunk, tiled reduction (JT=16).
// grid = (B, NC), block = H=512 threads (16 wave32s). Each block: (b, chunk c).
//   acc starts at exclusive chunk prefix; per j:
//     hid = sigmoid(acc + j*b_hidden[h]);  logit_j = sum_h hid*alpha[j,h] + ab[j]
//     ll += x*log(sig(logit)) + (1-x)*(1-log(sig(logit)));  acc += x[b,j]*W[h,j]
// Phase A computes 16 j's of wave partials barrier-free; Phase B finishes 16
// logits in parallel on threads 0..15. Two barriers per 16 j's (was 2 per j).
// ---------------------------------------------------------------------------
__global__ __launch_bounds__(NH)
void nade_seq(const float* __restrict__ x,
              const float* __restrict__ W,
              const float* __restrict__ bh,
              const float* __restrict__ aw,
              const float* __restrict__ ab,
              const float* __restrict__ P,   // exclusive chunk prefixes
              float* __restrict__ LL) {      // [B, NC] partials
    const int b = blockIdx.x;
    const int c = blockIdx.y;
    const int h = threadIdx.x;
    const int lane = h & 31;
    const int wid  = h >> 5;                 // 0..15

    __shared__ float red[JT][NH / 32];       // [16 j][16 wave partials]

    float acc = P[(size_t)c * NB * NH + (size_t)b * NH + h];
    const float bhv = bh[h];
    const float* __restrict__ wr = W + (size_t)h * ND;
    const int j0 = c * CS;
    float ll = 0.0f;                         // live on threads 0..15 only

    for (int tt = 0; tt < CS; tt += JT) {
        // ---- Phase A: 16 j-steps, barrier-free -------------------------
#pragma unroll
        for (int u = 0; u < JT; ++u) {
            const int j = j0 + tt + u;
            const float pre = fmaf((float)j, bhv, acc);
            const float hid = 1.0f / (1.0f + __expf(-pre));
            float part = hid * aw[(size_t)j * NH + h];
            part = wave_reduce_add(part);
            if (lane == 0) red[u][wid] = part;
            // recurrence update (x[b,j] wave-uniform -> scalarized)
            acc = fmaf(x[(size_t)b * ND + j], wr[j], acc);
        }
        __syncthreads();

        // ---- Phase B: finish 16 logits in parallel ---------------------
        if (h < JT) {
            const int j = j0 + tt + h;
            float logit = ab[j];
#pragma unroll
            for (int w = 0; w < NH / 32; ++w) logit += red[h][w];
            const float xj = x[(size_t)b * ND + j];
            const float lm = -log1pf(__expf(-logit));  // log(sigmoid(logit))
            ll += xj * lm + (1.0f - xj) * (1.0f - lm);
        }
        __syncthreads();  // protect red[] before next tile's writes
    }

    // combine per-j ll accumulators (threads 0..15 live in wave 0)
    if (wid == 0) {
        float v = (lane < JT) ? ll : 0.0f;
        v = wave_reduce_add(v);
        if (lane == 0) LL[b * NC + c] = v;
    }
}

// ---------------------------------------------------------------------------
// Kernel 4: out[b] = sum_c LL[b,c]   (re-initializes poisoned d_out each call)
// ---------------------------------------------------------------------------
__global__ void nade_final(const float* __restrict__ LL, float* __restrict__ out) {
    const int b = blockIdx.x * blockDim.x + threadIdx.x;
    if (b < NB) {
        float s = 0.0f;
#pragma unroll
        for (int c = 0; c < NC; ++c) s += LL[b * NC + c];
        out[b] = s;
    }
}

// ---------------------------------------------------------------------------
// Launcher. Scratch layout in d_ws (floats):
//   [0, NC*NB*NH)        chunk sums -> exclusive prefixes (2 MB)
//   [NC*NB*NH, +NB*NC)   per-(b,chunk) log-lik partials   (4 KB)
// ---------------------------------------------------------------------------
extern "C" void kernel_launch(void* const* d_in, const int* in_sizes, int n_in,
                              void* d_out, int out_size, void* d_ws, size_t ws_size,
                              hipStream_t stream) {
    const float* x  = (const float*)d_in[0];  // [B, D]
    const float* W  = (const float*)d_in[1];  // [H, D]
    const float* bh = (const float*)d_in[2];  // [H]
    const float* aw = (const float*)d_in[3];  // [D, H]
    const float* ab = (const float*)d_in[4];  // [D]
    float* out = (float*)d_out;               // [B]

    float* S  = (float*)d_ws;
    float* LL = S + (size_t)NC * NB * NH;

    // 1) chunk partial-sum GEMM via WMMA: 2048 tiles, 8 waves/block
    const int totalTiles = NC * (NB / 16) * (NH / 16);
    nade_chunk_gemm<<<totalTiles / 8, 256, 0, stream>>>(x, W, S);

    // 2) exclusive scan over chunks per (b,h)
    nade_prefix<<<(NB * NH) / 256, 256, 0, stream>>>(S);

    // 3) sequential recurrence per (b, chunk)
    dim3 g3(NB, NC);
    nade_seq<<<g3, NH, 0, stream>>>(x, W, bh, aw, ab, S, LL);

    // 4) final reduction
    nade_final<<<1, NB, 0, stream>>>(LL, out);
}